// GANLoss_64450279243851
// MI455X (gfx1250) — compile-verified
//
#include <hip/hip_runtime.h>
#include <hip/hip_bf16.h>

// Reference: out = -mean_n( reward[n] * mean_f( prob_flat[n, OFF[f] + target[n,f]] ) )
// N = 65536 rows, C = 3200 columns, F = 6 fields. Output: single f32 scalar.
//
// Structure:
//   Kernel 1: 1 thread per row -> 6 gathers + reward scale, deterministic
//             LDS tree reduce (256 -> 32), then exact f32 WMMA wave reduce
//             (V_WMMA_F32_16X16X4_F32) -> one partial per block in d_ws.
//   Kernel 2: single wave32 reduces the per-block partials (fixed order) with
//             the same WMMA wave reduction, writes -sum/N.
// No atomics -> bit-deterministic across graph replays.

typedef __attribute__((ext_vector_type(2))) float v2f;
typedef __attribute__((ext_vector_type(8))) float v8f;

__device__ __forceinline__ float wave32_sum_wmma(float v) {
    // A (16x4 f32): lanes 0-15 hold K=0 (VGPR0) / K=1 (VGPR1); lanes 16-31 hold K=2/3.
    // a = {v, 0} => row m of A sums to p_m + p_{m+16}.
    v2f a; a.x = v;    a.y = 0.0f;
    v2f b; b.x = 1.0f; b.y = 1.0f;   // B (4x16) all ones, layout-independent
    v8f c = {};
    // D[m][n] = sum_k A[m][k]  (identical for every column n)
    c = __builtin_amdgcn_wmma_f32_16x16x4_f32(
        /*neg_a=*/false, a, /*neg_b=*/false, b,
        /*c_mod=*/(short)0, c, /*reuse_a=*/false, /*reuse_b=*/false);
    // Lanes 0-15: VGPR v = D[v][lane]   -> sum over v gives rows 0..7
    // Lanes 16-31: VGPR v = D[v+8][..]  -> sum over v gives rows 8..15
    float s = c[0] + c[1] + c[2] + c[3] + c[4] + c[5] + c[6] + c[7];
    s += __shfl_xor(s, 16, 32);      // combine the two half-sums -> total on all lanes
    return s;
}

__global__ __launch_bounds__(256)
void gan_gather_partials(const float* __restrict__ prob,
                         const long long* __restrict__ target,
                         const float* __restrict__ reward,
                         float* __restrict__ partials,
                         int N) {
    __shared__ float sdata[256];
    const int tid = threadIdx.x;
    const int n   = blockIdx.x * 256 + tid;

    float s = 0.0f;
    if (n < N) {
        const long long  nl  = (long long)n;
        const float*     row = prob + nl * 3200;
        const long long* t   = target + nl * 6;
        // Field offsets: cumsum([256,256,1024,1024,512,128]) -> [0,256,512,1536,2560,3072]
        long long t0 = t[0], t1 = t[1], t2 = t[2], t3 = t[3], t4 = t[4], t5 = t[5];
        float g0 = row[   0 + (int)t0];
        float g1 = row[ 256 + (int)t1];
        float g2 = row[ 512 + (int)t2];
        float g3 = row[1536 + (int)t3];
        float g4 = row[2560 + (int)t4];
        float g5 = row[3072 + (int)t5];
        s = (g0 + g1 + g2 + g3 + g4 + g5) * (1.0f / 6.0f) * reward[n];
    }

    sdata[tid] = s;
    __syncthreads();
    // Deterministic tree: 256 -> 32
    #pragma unroll
    for (int str = 128; str >= 32; str >>= 1) {
        if (tid < str) sdata[tid] += sdata[tid + str];
        __syncthreads();
    }
    if (tid < 32) {                      // uniform branch: wave 0 has EXEC all-1s
        float tot = wave32_sum_wmma(sdata[tid]);
        if (tid == 0) partials[blockIdx.x] = tot;
    }
}

__global__ __launch_bounds__(32)
void gan_finalize(const float* __restrict__ partials,
                  float* __restrict__ out,
                  int nPart, float invN) {
    const int tid = threadIdx.x;         // exactly one wave32, EXEC all-1s
    float v = 0.0f;
    for (int i = tid; i < nPart; i += 32) v += partials[i];  // fixed order per lane
    float tot = wave32_sum_wmma(v);
    if (tid == 0) out[0] = -tot * invN;
}

extern "C" void kernel_launch(void* const* d_in, const int* in_sizes, int n_in,
                              void* d_out, int out_size, void* d_ws, size_t ws_size,
                              hipStream_t stream) {
    const float*     prob   = (const float*)d_in[0];      // (N, 3200) f32
    const long long* target = (const long long*)d_in[1];  // (N, 6) i64
    const float*     reward = (const float*)d_in[2];      // (N,) f32
    // d_in[3] = weights (unused by the reference forward)

    const int N       = in_sizes[2];                      // 65536
    const int nBlocks = (N + 255) / 256;                  // 256

    float* partials = (float*)d_ws;                       // nBlocks floats, fully
                                                          // overwritten every call
    float* out      = (float*)d_out;

    gan_gather_partials<<<nBlocks, 256, 0, stream>>>(prob, target, reward, partials, N);
    gan_finalize<<<1, 32, 0, stream>>>(partials, out, nBlocks, 1.0f / (float)N);
}